// TemporalAttention_63771674411180
// MI455X (gfx1250) — compile-verified
//
#include <hip/hip_runtime.h>
#include <math.h>

typedef float v2f __attribute__((ext_vector_type(2)));
typedef float v8f __attribute__((ext_vector_type(8)));

#define B_   16
#define C_IN 16
#define N_   512
#define T_   64
#define D_   8
#define CNT  ((float)(B_ * T_ * N_))

// workspace layout (in floats)
#define OFF_YQ  0ull
#define OFF_YK  (OFF_YQ + (size_t)B_ * T_ * N_ * D_)
#define OFF_YV  (OFF_YK + (size_t)B_ * T_ * N_ * D_)
#define OFF_YO  (OFF_YV + (size_t)B_ * T_ * N_ * D_)
#define OFF_RAW (OFF_YO + (size_t)B_ * T_ * N_ * 2 * D_)
#define OFF_FIN (OFF_RAW + 80)
// RAW: [0:8) sum_q [8:16) ss_q [16:24) sum_k [24:32) ss_k [32:40) sum_v [40:48) ss_v
//      [48:64) sum_o [64:80) ss_o
// FIN: same slots but scale/shift

// ---------------------------------------------------------------------------
// Kernel 1: y{q,k,v} = xt @ W + b  via f32 WMMA, K=16 as 4 chained K=4 WMMAs.
// One wave handles a 16-row tile (rows = 16 consecutive n at fixed b,t).
// B operand packs Wq (cols 0-7) | Wk (cols 8-15); second chain does Wv.
// Also accumulates per-channel sum / sum-of-squares for BatchNorm.
// ---------------------------------------------------------------------------
__global__ __launch_bounds__(32)
void fc_qkv_kernel(const float* __restrict__ x,
                   const float* __restrict__ Wq, const float* __restrict__ bq,
                   const float* __restrict__ Wk, const float* __restrict__ bk,
                   const float* __restrict__ Wv, const float* __restrict__ bv,
                   float* __restrict__ yq, float* __restrict__ yk,
                   float* __restrict__ yv, float* __restrict__ raw)
{
    const int lane = threadIdx.x;
    const int col  = lane & 15;    // A row index AND output column index
    const int hi   = lane >> 4;

    const int nt   = N_ / 16;
    const int tile = blockIdx.x;            // B*T*(N/16) tiles
    const int n0   = (tile % nt) * 16;
    const int bt   = tile / nt;
    const int b    = bt / T_;
    const int t    = bt % T_;

    // xt[b,t,n,c] = x[b,c,n,t];  x index = ((b*Cin + c)*N + n)*T + t
    const long xbase = (long)b * C_IN * N_ * T_ + (long)(n0 + col) * T_ + t;

    const float biasqk = (col < 8) ? bq[col] : bk[col - 8];
    const float biasv  = (col < 8) ? bv[col] : 0.0f;
    v8f accqk, accv;
    for (int v = 0; v < 8; ++v) { accqk[v] = biasqk; accv[v] = biasv; }

    for (int kb = 0; kb < 4; ++kb) {
        const int k0 = 4 * kb + (hi ? 2 : 0);   // A layout: lanes 0-15 K={0,1}, 16-31 K={2,3}
        v2f a, bqk, bvv;
        a.x = x[xbase + (long)k0 * (N_ * T_)];
        a.y = x[xbase + (long)(k0 + 1) * (N_ * T_)];
        if (col < 8) {
            bqk.x = Wq[k0 * D_ + col];       bqk.y = Wq[(k0 + 1) * D_ + col];
            bvv.x = Wv[k0 * D_ + col];       bvv.y = Wv[(k0 + 1) * D_ + col];
        } else {
            bqk.x = Wk[k0 * D_ + (col - 8)]; bqk.y = Wk[(k0 + 1) * D_ + (col - 8)];
            bvv.x = 0.0f;                    bvv.y = 0.0f;
        }
        accqk = __builtin_amdgcn_wmma_f32_16x16x4_f32(false, a, false, bqk, (short)0, accqk, false, false);
        accv  = __builtin_amdgcn_wmma_f32_16x16x4_f32(false, a, false, bvv, (short)0, accv,  false, false);
    }

    // Store + BN raw stats. C/D layout: lane holds rows v + hi*8, column `col`.
    float s1 = 0.0f, s2 = 0.0f, sv1 = 0.0f, sv2 = 0.0f;
    const long rowbase = (long)(b * T_ + t) * N_ + n0;
    for (int v = 0; v < 8; ++v) {
        const int  rM  = v + hi * 8;
        const long r   = rowbase + rM;
        const float dq = accqk[v];
        const float dv = accv[v];
        if (col < 8) {
            yq[r * D_ + col] = dq;
            yv[r * D_ + col] = dv;
            sv1 += dv; sv2 += dv * dv;
        } else {
            yk[r * D_ + (col - 8)] = dq;
        }
        s1 += dq; s2 += dq * dq;
    }
    s1  += __shfl_xor(s1, 16, 32);  s2  += __shfl_xor(s2, 16, 32);
    sv1 += __shfl_xor(sv1, 16, 32); sv2 += __shfl_xor(sv2, 16, 32);
    if (hi == 0) {
        if (col < 8) {
            atomicAdd(&raw[0  + col], s1);
            atomicAdd(&raw[8  + col], s2);
            atomicAdd(&raw[32 + col], sv1);
            atomicAdd(&raw[40 + col], sv2);
        } else {
            atomicAdd(&raw[16 + (col - 8)], s1);
            atomicAdd(&raw[24 + (col - 8)], s2);
        }
    }
}

// ---------------------------------------------------------------------------
// Kernel 2: fold q/k/v batch stats into fused BN scale/shift per channel.
// ---------------------------------------------------------------------------
__global__ __launch_bounds__(32)
void stats_qkv_kernel(const float* __restrict__ raw, float* __restrict__ fin,
                      const float* gq, const float* betaq,
                      const float* gk, const float* betak,
                      const float* gv, const float* betav)
{
    const int tid = threadIdx.x;
    if (tid >= 24) return;
    const int grp = tid >> 3;   // 0=q 1=k 2=v
    const int c   = tid & 7;
    const float* g;  const float* be;
    if (grp == 0)      { g = gq; be = betaq; }
    else if (grp == 1) { g = gk; be = betak; }
    else               { g = gv; be = betav; }
    const float mean = raw[grp * 16 + c] / CNT;
    const float var  = raw[grp * 16 + 8 + c] / CNT - mean * mean;
    const float rs   = rsqrtf(var + 1e-5f);
    const float sc   = g[c] * rs;
    fin[grp * 16 + c]     = sc;
    fin[grp * 16 + 8 + c] = be[c] - mean * sc;
}

// ---------------------------------------------------------------------------
// Kernel 3: per-(b,n) temporal attention. 4 waves = 4 heads.
// V tile staged into LDS with gfx1250 async LDS-DMA (ASYNCcnt-tracked),
// scores 64x64 per head as 4x4 WMMA tiles (K=2 packed in a K=4 f32 WMMA),
// softmax in the C-register layout with half-wave shuffles, attn@V by VALU,
// then out@Wo + bo, with o-channel BN stats accumulated.
// ---------------------------------------------------------------------------
__global__ __launch_bounds__(128)
void attn_kernel(const float* __restrict__ yq, const float* __restrict__ yk,
                 const float* __restrict__ yv, const float* __restrict__ Wo,
                 const float* __restrict__ bo, const float* __restrict__ fin,
                 float* __restrict__ yo, float* __restrict__ raw)
{
    __shared__ float vn[T_][D_];     // rows are 32B-contiguous -> 2x b128 per row
    __shared__ float outh[T_][D_];
    __shared__ float red[128];

    const int site = blockIdx.x;
    const int b    = site / N_;
    const int n    = site % N_;
    const int tid  = threadIdx.x;
    const int h    = tid >> 5;     // head = wave id
    const int lane = tid & 31;
    const int col  = lane & 15;
    const int hi   = lane >> 4;

    // ---- async LDS-DMA: raw V tile (64 rows x 32B) as 128 x b128 transfers
    {
        const int t = tid >> 1, half = tid & 1;
        const float* gp = &yv[((long)(b * T_ + t) * N_ + n) * D_ + half * 4];
        const unsigned ldsoff =
            (unsigned)(unsigned long long)(const void*)&vn[t][half * 4];
        asm volatile("global_load_async_to_lds_b128 %0, %1, off"
                     :: "v"(ldsoff), "v"((unsigned long long)gp)
                     : "memory");
    }

    // ---- prefetch the Q/K rows this block will gather below
    {
        const int pt = tid & 63;
        const float* pp = (tid < 64)
            ? &yq[((long)(b * T_ + pt) * N_ + n) * D_]
            : &yk[((long)(b * T_ + pt) * N_ + n) * D_];
        __builtin_prefetch(pp, 0, 1);
    }

    // wait for this wave's async transfers, then barrier so all rows are visible
    asm volatile("s_wait_asynccnt 0x0" ::: "memory");
    __syncthreads();

    // normalize V in place (BN + ReLU)
    for (int i = 0; i < 4; ++i) {
        const int idx = tid + i * 128;
        const int t = idx >> 3, c = idx & 7;
        vn[t][c] = fmaxf(vn[t][c] * fin[32 + c] + fin[40 + c], 0.0f);
    }
    __syncthreads();

    const int c0 = 2 * h, c1 = 2 * h + 1;
    const float sq0 = fin[c0],      sq1 = fin[c1];
    const float hq0 = fin[8 + c0],  hq1 = fin[8 + c1];
    const float sk0 = fin[16 + c0], sk1 = fin[16 + c1];
    const float hk0 = fin[24 + c0], hk1 = fin[24 + c1];

    // A = Q tiles (16 x K, K=2 real + 2 zero), B = K^T tiles (K x 16)
    v2f A[4], Bm[4];
    for (int ti = 0; ti < 4; ++ti) {
        if (hi == 0) {
            const int t = ti * 16 + col;
            const long rb = ((long)(b * T_ + t) * N_ + n) * D_;
            A[ti].x = fmaxf(yq[rb + c0] * sq0 + hq0, 0.0f);
            A[ti].y = fmaxf(yq[rb + c1] * sq1 + hq1, 0.0f);
        } else { A[ti].x = 0.0f; A[ti].y = 0.0f; }
    }
    for (int tj = 0; tj < 4; ++tj) {
        if (hi == 0) {
            const int t = tj * 16 + col;
            const long rb = ((long)(b * T_ + t) * N_ + n) * D_;
            Bm[tj].x = fmaxf(yk[rb + c0] * sk0 + hk0, 0.0f);
            Bm[tj].y = fmaxf(yk[rb + c1] * sk1 + hk1, 0.0f);
        } else { Bm[tj].x = 0.0f; Bm[tj].y = 0.0f; }
    }

    v8f S[4][4];
    for (int ti = 0; ti < 4; ++ti)
        for (int tj = 0; tj < 4; ++tj) {
            v8f z; for (int v = 0; v < 8; ++v) z[v] = 0.0f;
            S[ti][tj] = __builtin_amdgcn_wmma_f32_16x16x4_f32(false, A[ti], false, Bm[tj], (short)0, z, false, false);
        }

    // softmax over full rows (scale 1/sqrt(4) folded into exp)
    for (int ti = 0; ti < 4; ++ti) {
        for (int v = 0; v < 8; ++v) {
            float m = S[ti][0][v];
            m = fmaxf(m, S[ti][1][v]); m = fmaxf(m, S[ti][2][v]); m = fmaxf(m, S[ti][3][v]);
            for (int off = 1; off < 16; off <<= 1) m = fmaxf(m, __shfl_xor(m, off, 32));
            float sum = 0.0f;
            for (int tj = 0; tj < 4; ++tj) {
                const float e = __expf(0.5f * (S[ti][tj][v] - m));
                S[ti][tj][v] = e; sum += e;
            }
            for (int off = 1; off < 16; off <<= 1) sum += __shfl_xor(sum, off, 32);
            const float inv = 1.0f / sum;
            for (int tj = 0; tj < 4; ++tj) S[ti][tj][v] *= inv;
        }
    }

    // out[t][2h+c] = sum_t' attn[t][t'] * vn[t'][2h+c]
    for (int ti = 0; ti < 4; ++ti) {
        for (int v = 0; v < 8; ++v) {
            float p0 = 0.0f, p1 = 0.0f;
            for (int tj = 0; tj < 4; ++tj) {
                const float a = S[ti][tj][v];
                p0 += a * vn[tj * 16 + col][c0];
                p1 += a * vn[tj * 16 + col][c1];
            }
            for (int off = 1; off < 16; off <<= 1) {
                p0 += __shfl_xor(p0, off, 32);
                p1 += __shfl_xor(p1, off, 32);
            }
            if (col == 0) {
                const int t = ti * 16 + v + hi * 8;
                outh[t][c0] = p0; outh[t][c1] = p1;
            }
        }
    }
    __syncthreads();

    // yo = out @ Wo + bo  (8 -> 16), plus o-channel BN raw stats
    float s1 = 0.0f, s2 = 0.0f;
    const int d = tid & 15;
    for (int i = 0; i < 8; ++i) {
        const int e = tid + i * 128;
        const int t = e >> 4;
        float acc = bo[d];
        for (int c = 0; c < 8; ++c) acc += outh[t][c] * Wo[c * 16 + d];
        yo[((long)(b * T_ + t) * N_ + n) * (2 * D_) + d] = acc;
        s1 += acc; s2 += acc * acc;
    }
    red[tid] = s1;
    __syncthreads();
    if (tid < 16) {
        float a = 0.0f; for (int j = 0; j < 8; ++j) a += red[tid + j * 16];
        atomicAdd(&raw[48 + tid], a);
    }
    __syncthreads();
    red[tid] = s2;
    __syncthreads();
    if (tid < 16) {
        float a = 0.0f; for (int j = 0; j < 8; ++j) a += red[tid + j * 16];
        atomicAdd(&raw[64 + tid], a);
    }
}

// ---------------------------------------------------------------------------
// Kernel 4: fold o-channel batch stats into scale/shift.
// ---------------------------------------------------------------------------
__global__ __launch_bounds__(32)
void stats_o_kernel(const float* __restrict__ raw, float* __restrict__ fin,
                    const float* go, const float* betao)
{
    const int c = threadIdx.x;
    if (c >= 16) return;
    const float mean = raw[48 + c] / CNT;
    const float var  = raw[64 + c] / CNT - mean * mean;
    const float rs   = rsqrtf(var + 1e-5f);
    const float sc   = go[c] * rs;
    fin[48 + c] = sc;
    fin[64 + c] = betao[c] - mean * sc;
}

// ---------------------------------------------------------------------------
// Kernel 5: BN + ReLU + transpose to [B, 16, N, T] via LDS tile so both the
// yo reads (d-contiguous, 64B rows) and the out writes (t-contiguous, 256B
// runs) are coalesced. One block per (b, n).
// ---------------------------------------------------------------------------
__global__ __launch_bounds__(256)
void apply_o_kernel(const float* __restrict__ yo, const float* __restrict__ fin,
                    float* __restrict__ out)
{
    __shared__ float tile[T_][2 * D_ + 1];   // +1 pad to spread LDS banks

    const int site = blockIdx.x;             // B*N blocks
    const int b = site / N_;
    const int n = site % N_;
    const int tid = threadIdx.x;

    for (int i = 0; i < 4; ++i) {
        const int e = tid + i * 256;         // 1024 elements
        const int t = e >> 4, d = e & 15;
        tile[t][d] = yo[((long)(b * T_ + t) * N_ + n) * (2 * D_) + d];
    }
    __syncthreads();
    for (int i = 0; i < 4; ++i) {
        const int e = tid + i * 256;
        const int d = e >> 6, t = e & 63;
        const float v = tile[t][d];
        out[((long)(b * (2 * D_) + d) * N_ + n) * (long)T_ + t] =
            fmaxf(v * fin[48 + d] + fin[64 + d], 0.0f);
    }
}

// ---------------------------------------------------------------------------
extern "C" void kernel_launch(void* const* d_in, const int* in_sizes, int n_in,
                              void* d_out, int out_size, void* d_ws, size_t ws_size,
                              hipStream_t stream)
{
    const float* x     = (const float*)d_in[0];
    // d_in[1] = time_ind (unused by the math)
    const float* Wq    = (const float*)d_in[2];
    const float* bq    = (const float*)d_in[3];
    const float* gq    = (const float*)d_in[4];
    const float* betaq = (const float*)d_in[5];
    const float* Wk    = (const float*)d_in[6];
    const float* bk    = (const float*)d_in[7];
    const float* gk    = (const float*)d_in[8];
    const float* betak = (const float*)d_in[9];
    const float* Wv    = (const float*)d_in[10];
    const float* bv    = (const float*)d_in[11];
    const float* gv    = (const float*)d_in[12];
    const float* betav = (const float*)d_in[13];
    const float* Wo    = (const float*)d_in[14];
    const float* bo    = (const float*)d_in[15];
    const float* go    = (const float*)d_in[16];
    const float* betao = (const float*)d_in[17];

    float* ws  = (float*)d_ws;
    float* yq  = ws + OFF_YQ;
    float* yk  = ws + OFF_YK;
    float* yv  = ws + OFF_YV;
    float* yo  = ws + OFF_YO;
    float* raw = ws + OFF_RAW;
    float* fin = ws + OFF_FIN;

    hipMemsetAsync(raw, 0, 80 * sizeof(float), stream);

    fc_qkv_kernel<<<B_ * T_ * (N_ / 16), 32, 0, stream>>>(
        x, Wq, bq, Wk, bk, Wv, bv, yq, yk, yv, raw);

    stats_qkv_kernel<<<1, 32, 0, stream>>>(raw, fin, gq, betaq, gk, betak, gv, betav);

    attn_kernel<<<B_ * N_, 128, 0, stream>>>(yq, yk, yv, Wo, bo, fin, yo, raw);

    stats_o_kernel<<<1, 32, 0, stream>>>(raw, fin, go, betao);

    apply_o_kernel<<<(B_ * N_), 256, 0, stream>>>(yo, fin, (float*)d_out);
}